// BranchGCN_3951369912528
// MI455X (gfx1250) — compile-verified
//
#include <hip/hip_runtime.h>
#include <hip/hip_bf16.h>
#include <cstddef>

typedef float v2f __attribute__((ext_vector_type(2)));
typedef float v8f __attribute__((ext_vector_type(8)));

#define BB    16
#define NODE  32
#define DEG   64
#define KNN   8
#define IN_F  128
#define OUT_F 3
#define SUP   10
#define NPTS  2048   // NODE*DEG
#define OCOLS (DEG * IN_F)   // 8192

// ---------------------------------------------------------------------------
// K1: fuse Wl1@Wl2 -> WlW (128x3);  c2w@c1w -> M3 (3x6);  fused bias (3)
// ---------------------------------------------------------------------------
__global__ __launch_bounds__(256)
void fuse_weights_kernel(const float* __restrict__ Wl1,   // (128, 1280)
                         const float* __restrict__ Wl2,   // (1280, 3)
                         const float* __restrict__ c1w,   // (64, 6)
                         const float* __restrict__ c1b,   // (64)
                         const float* __restrict__ c2w,   // (3, 64)
                         const float* __restrict__ c2b,   // (3)
                         float* __restrict__ WlW,         // (128, 3)
                         float* __restrict__ M3,          // (3, 6)
                         float* __restrict__ bf) {        // (3)
  const int t = blockIdx.x * blockDim.x + threadIdx.x;
  if (t < IN_F * OUT_F) {
    const int i = t / OUT_F, c = t % OUT_F;
    float s = 0.f;
    for (int k = 0; k < IN_F * SUP; ++k)
      s += Wl1[(size_t)i * (IN_F * SUP) + k] * Wl2[k * OUT_F + c];
    WlW[t] = s;
  }
  if (t < 18) {
    const int c = t / 6, j = t % 6;
    float s = 0.f;
    for (int o = 0; o < 64; ++o) s += c2w[c * 64 + o] * c1w[o * 6 + j];
    M3[t] = s;
  }
  if (t < 3) {
    float s = c2b[t];
    for (int o = 0; o < 64; ++o) s += c2w[t * 64 + o] * c1b[o];
    bf[t] = s;
  }
}

// ---------------------------------------------------------------------------
// K2: root[b,n,c] = sum_i repeat(t_i @ Wr_i, 32/rn_i)[b,n,c]
// ---------------------------------------------------------------------------
__global__ __launch_bounds__(256)
void root_kernel(const float* __restrict__ t0, const float* __restrict__ t1,
                 const float* __restrict__ t2, const float* __restrict__ t3,
                 const float* __restrict__ t4, const float* __restrict__ t5,
                 const float* __restrict__ W0, const float* __restrict__ W1,
                 const float* __restrict__ W2, const float* __restrict__ W3,
                 const float* __restrict__ W4, const float* __restrict__ W5,
                 float* __restrict__ root) {
  const int t = blockIdx.x * blockDim.x + threadIdx.x;
  if (t >= BB * NODE * OUT_F) return;
  const int c = t % OUT_F;
  const int n = (t / OUT_F) % NODE;
  const int b = t / (NODE * OUT_F);
  float s = 0.f;
#define TREE_ACC(TP, WP, RN, F)                                   \
  do {                                                            \
    const int ri = n / (NODE / (RN));                             \
    const float* row = (TP) + ((size_t)b * (RN) + ri) * (F);      \
    float a_ = 0.f;                                               \
    for (int f = 0; f < (F); ++f) a_ += row[f] * (WP)[f * 3 + c]; \
    s += a_;                                                      \
  } while (0)
  TREE_ACC(t0, W0, 1, 96);
  TREE_ACC(t1, W1, 2, 256);
  TREE_ACC(t2, W2, 4, 256);
  TREE_ACC(t3, W3, 8, 256);
  TREE_ACC(t4, W4, 16, 128);
  TREE_ACC(t5, W5, 32, 128);
#undef TREE_ACC
  root[t] = s;
}

// ---------------------------------------------------------------------------
// K3: per node n:  pre = t5[:,n,:] (16x128) @ W_branch[n] (128x8192) via
//     V_WMMA_F32_16X16X4_F32, software-pipelined: B operands for the next
//     16-step K-chunk are fetched (32-load clause) while the current chunk's
//     16 chained WMMAs execute.  Epilogue fuses leaky + (128->3) projection
//     (WlW) + root add via xor-shuffle reduction.  One block per node, 8
//     waves; wave w covers d = w + 8*dd.
// ---------------------------------------------------------------------------
__global__ __launch_bounds__(256)
void branch_wmma_kernel(const float* __restrict__ t5,   // (16, 32, 128)
                        const float* __restrict__ Wb,   // (32, 128, 8192)
                        const float* __restrict__ root, // (16, 32, 3)
                        const float* __restrict__ WlW,  // (128, 3)
                        float* __restrict__ y) {        // (16, 2048, 3)
  const int n    = blockIdx.x;          // node
  const int wave = threadIdx.x >> 5;    // 0..7
  const int lane = threadIdx.x & 31;
  const int hi   = lane >> 4;           // 0/1 half
  const int lo   = lane & 15;

  // A = t5[:, n, :] (M=16, K=128), split into two K-halves so all indexing
  // stays compile-time.  ISA 16x4 f32 layout:
  // lane l: VGPR0 = A[m=l%16][k=2*(l/16)+4*s], VGPR1 = same k+1.
  v2f a0[16], a1[16];
  const float* arow = t5 + ((size_t)lo * NODE + n) * IN_F;
#pragma unroll
  for (int s = 0; s < 16; ++s) {
    const int k0 = 4 * s + 2 * hi;
    a0[s].x = arow[k0];
    a0[s].y = arow[k0 + 1];
    a1[s].x = arow[k0 + 64];
    a1[s].y = arow[k0 + 65];
  }

  const float* wbase = Wb + (size_t)n * IN_F * OCOLS + lo;

  // chunk u in [0,128): tile tt=u>>1 (d = wave+8*(tt>>3), jt = tt&7),
  // half = u&1 (K range [64*half, 64*half+64) in steps of 4).
  auto load_chunk = [&](int u, v2f (&buf)[16]) {
    const int tt   = u >> 1;
    const int d    = wave + 8 * (tt >> 3);
    const int jt   = tt & 7;
    const int half = u & 1;
    const float* bcol = wbase + (size_t)d * IN_F + jt * 16;
#pragma unroll
    for (int s2 = 0; s2 < 16; ++s2) {
      const int k0 = 4 * (16 * half + s2) + 2 * hi;
      buf[s2].x = bcol[(size_t)k0 * OCOLS];
      buf[s2].y = bcol[(size_t)(k0 + 1) * OCOLS];
    }
  };

  v2f b0[16], b1[16];
  v8f c;
  float acc[8][3];
#pragma unroll
  for (int r = 0; r < 8; ++r) { acc[r][0] = 0.f; acc[r][1] = 0.f; acc[r][2] = 0.f; }

  load_chunk(0, b0);

  for (int u = 0; u < 128; u += 2) {
    // ---- prefetch chunk u+1 while computing chunk u (K half 0) ----
    load_chunk(u + 1, b1);
    {
      v8f cc = {0.f, 0.f, 0.f, 0.f, 0.f, 0.f, 0.f, 0.f};
#pragma unroll
      for (int s2 = 0; s2 < 16; ++s2)
        cc = __builtin_amdgcn_wmma_f32_16x16x4_f32(
                 false, a0[s2], false, b0[s2], (short)0, cc, false, false);
      c = cc;
    }
    // ---- prefetch chunk u+2 while computing chunk u+1 (K half 1) ----
    if (u < 126) load_chunk(u + 2, b0);
    {
#pragma unroll
      for (int s2 = 0; s2 < 16; ++s2)
        c = __builtin_amdgcn_wmma_f32_16x16x4_f32(
                false, a1[s2], false, b1[s2], (short)0, c, false, false);
    }
    // ---- tile epilogue: leaky + project feature (jt*16+lo) onto 3 chans ----
    {
      const int tt = u >> 1;
      const int jt = tt & 7;
      const int ig = jt * 16 + lo;
      const float w0 = WlW[ig * 3 + 0];
      const float w1 = WlW[ig * 3 + 1];
      const float w2 = WlW[ig * 3 + 2];
#pragma unroll
      for (int r = 0; r < 8; ++r) {
        float v = c[r];
        v = (v >= 0.f) ? v : 0.2f * v;  // leaky on the einsum output
        acc[r][0] += v * w0;
        acc[r][1] += v * w1;
        acc[r][2] += v * w2;
      }
      // ---- d-block epilogue every 8 tiles: cross-lane reduce + store ----
      if ((u & 15) == 14) {
        const int d = wave + 8 * (tt >> 3);
#pragma unroll
        for (int r = 0; r < 8; ++r) {
#pragma unroll
          for (int ch = 0; ch < 3; ++ch) {
            float v = acc[r][ch];
            v += __shfl_xor(v, 1, 32);
            v += __shfl_xor(v, 2, 32);
            v += __shfl_xor(v, 4, 32);
            v += __shfl_xor(v, 8, 32);
            acc[r][ch] = v;
          }
        }
        if (lo == 0) {                  // lane 0 -> m=r, lane 16 -> m=r+8
          const int row = n * DEG + d;
#pragma unroll
          for (int r = 0; r < 8; ++r) {
            const int m = r + 8 * hi;   // batch
#pragma unroll
            for (int ch = 0; ch < 3; ++ch)
              y[((size_t)m * NPTS + row) * 3 + ch] =
                  root[(m * NODE + n) * 3 + ch] + acc[r][ch];
          }
        }
#pragma unroll
        for (int r = 0; r < 8; ++r) { acc[r][0] = 0.f; acc[r][1] = 0.f; acc[r][2] = 0.f; }
      }
    }
  }
}

// ---------------------------------------------------------------------------
// K4: per batch b (one block): load x (2048x3) into LDS, per point n find
//     top-8 neighbors of pd = 2<xn,xm> - |xn|^2 - |xm|^2 (jax top_k
//     tie-break preserved: strict '>' insertion, ascending m scan), then
//     fused edge-conv: out[c] = leaky(max_k(g . M3[c]) + bf[c] + bias[n%64,c])
// ---------------------------------------------------------------------------
__global__ __launch_bounds__(256)
void knn_edge_kernel(const float* __restrict__ y,    // (16, 2048, 3)
                     const float* __restrict__ M3,   // (3, 6)
                     const float* __restrict__ bf,   // (3)
                     const float* __restrict__ bias, // (1, 64, 3)
                     float* __restrict__ out) {      // (16, 2048, 3)
  __shared__ float sx[NPTS * 3];
  __shared__ float sxx[NPTS];
  const int b = blockIdx.x;
  const float* xb = y + (size_t)b * NPTS * 3;
  for (int t = threadIdx.x; t < NPTS * 3; t += blockDim.x) sx[t] = xb[t];
  __syncthreads();
  for (int t = threadIdx.x; t < NPTS; t += blockDim.x) {
    const float x0 = sx[3 * t], x1 = sx[3 * t + 1], x2 = sx[3 * t + 2];
    sxx[t] = x0 * x0 + x1 * x1 + x2 * x2;
  }
  __syncthreads();

  float m3[18];
#pragma unroll
  for (int i = 0; i < 18; ++i) m3[i] = M3[i];
  const float bf0 = bf[0], bf1 = bf[1], bf2 = bf[2];

  for (int n = threadIdx.x; n < NPTS; n += blockDim.x) {
    const float xn0 = sx[3 * n], xn1 = sx[3 * n + 1], xn2 = sx[3 * n + 2];
    const float xxn = sxx[n];
    float val[KNN];
    int   idx[KNN];
#pragma unroll
    for (int k = 0; k < KNN; ++k) { val[k] = -3.0e38f; idx[k] = 0; }

    for (int m = 0; m < NPTS; ++m) {
      const float pd = 2.f * (xn0 * sx[3 * m] + xn1 * sx[3 * m + 1] +
                              xn2 * sx[3 * m + 2]) - xxn - sxx[m];
      if (pd > val[KNN - 1]) {
        val[KNN - 1] = pd;
        idx[KNN - 1] = m;
#pragma unroll
        for (int j = KNN - 1; j > 0; --j) {
          if (val[j] > val[j - 1]) {
            const float tv = val[j]; val[j] = val[j - 1]; val[j - 1] = tv;
            const int   ti = idx[j]; idx[j] = idx[j - 1]; idx[j - 1] = ti;
          }
        }
      }
    }

    float best0 = -3.0e38f, best1 = -3.0e38f, best2 = -3.0e38f;
#pragma unroll
    for (int k = 0; k < KNN; ++k) {
      const int mm = idx[k];
      const float f0 = sx[3 * mm]     - xn0;
      const float f1 = sx[3 * mm + 1] - xn1;
      const float f2 = sx[3 * mm + 2] - xn2;
      const float h0 = f0 * m3[0]  + f1 * m3[1]  + f2 * m3[2] +
                       xn0 * m3[3] + xn1 * m3[4] + xn2 * m3[5] + bf0;
      const float h1 = f0 * m3[6]  + f1 * m3[7]  + f2 * m3[8] +
                       xn0 * m3[9] + xn1 * m3[10] + xn2 * m3[11] + bf1;
      const float h2 = f0 * m3[12] + f1 * m3[13] + f2 * m3[14] +
                       xn0 * m3[15] + xn1 * m3[16] + xn2 * m3[17] + bf2;
      best0 = fmaxf(best0, h0);
      best1 = fmaxf(best1, h1);
      best2 = fmaxf(best2, h2);
    }
    const int dpos = n & (DEG - 1);
    float o0 = best0 + bias[dpos * 3 + 0];
    float o1 = best1 + bias[dpos * 3 + 1];
    float o2 = best2 + bias[dpos * 3 + 2];
    o0 = (o0 >= 0.f) ? o0 : 0.2f * o0;
    o1 = (o1 >= 0.f) ? o1 : 0.2f * o1;
    o2 = (o2 >= 0.f) ? o2 : 0.2f * o2;
    float* op = out + ((size_t)b * NPTS + n) * 3;
    op[0] = o0; op[1] = o1; op[2] = o2;
  }
}

// ---------------------------------------------------------------------------
// host side
// ---------------------------------------------------------------------------
extern "C" void kernel_launch(void* const* d_in, const int* in_sizes, int n_in,
                              void* d_out, int out_size, void* d_ws, size_t ws_size,
                              hipStream_t stream) {
  const float* t0  = (const float*)d_in[0];
  const float* t1  = (const float*)d_in[1];
  const float* t2  = (const float*)d_in[2];
  const float* t3  = (const float*)d_in[3];
  const float* t4  = (const float*)d_in[4];
  const float* t5  = (const float*)d_in[5];
  const float* Wr0 = (const float*)d_in[6];
  const float* Wr1 = (const float*)d_in[7];
  const float* Wr2 = (const float*)d_in[8];
  const float* Wr3 = (const float*)d_in[9];
  const float* Wr4 = (const float*)d_in[10];
  const float* Wr5 = (const float*)d_in[11];
  const float* Wb  = (const float*)d_in[12];
  const float* Wl1 = (const float*)d_in[13];
  const float* Wl2 = (const float*)d_in[14];
  const float* bias= (const float*)d_in[15];
  const float* c1w = (const float*)d_in[16];
  const float* c1b = (const float*)d_in[17];
  const float* c2w = (const float*)d_in[18];
  const float* c2b = (const float*)d_in[19];

  float* ws   = (float*)d_ws;
  float* y    = ws;                     // 16*2048*3 = 98304
  float* root = ws + 98304;             // 1536
  float* WlW  = ws + 98304 + 1536;      // 384
  float* M3   = ws + 98304 + 1536 + 384;// 18
  float* bf   = ws + 98304 + 1536 + 384 + 18; // 3
  float* outp = (float*)d_out;

  fuse_weights_kernel<<<2, 256, 0, stream>>>(Wl1, Wl2, c1w, c1b, c2w, c2b,
                                             WlW, M3, bf);
  root_kernel<<<6, 256, 0, stream>>>(t0, t1, t2, t3, t4, t5,
                                     Wr0, Wr1, Wr2, Wr3, Wr4, Wr5, root);
  branch_wmma_kernel<<<NODE, 256, 0, stream>>>(t5, Wb, root, WlW, y);
  knn_edge_kernel<<<BB, 256, 0, stream>>>(y, M3, bf, bias, outp);
}